// FFTFeatureExtractor_52518860095927
// MI455X (gfx1250) — compile-verified
//
#include <hip/hip_runtime.h>
#include <math.h>

// ---------------------------------------------------------------------------
// FFTFeatureExtractor for MI455X (gfx1250, wave32).
// Pipeline (all on `stream`):
//   K1: gray conversion fused with per-row 512-pt FFT (1 wave32 per row,
//       wave-synchronous LDS butterflies, no barriers needed inside FFT).
//   K2: per-column 512-pt FFT (1 wave32 per column, 8 cols/block), fused
//       magnitude + fftshift + deterministic per-block partial reductions.
//   K3: per-image finalize: threshold counts + 4-level histogram-refinement
//       median over the L2-resident |F| field, emit 12 features.
//   K4: MLP 12->64->128->512 with fp32 WMMA (v_wmma_f32_16x16x4_f32),
//       training-mode BatchNorm (batch stats, biased var) + ReLU in LDS.
// Workspace requirement: ~202 MB (rowfft 134MB + mag 67MB + partials).
// ---------------------------------------------------------------------------

#define BATCH 64
#define HH    512
#define WW    512
#define HWSZ  (HH * WW)          // 262144
#define NTOT  ((float)HWSZ)
#define MED_K 131071             // (n-1)//2, 0-based lower-median index

typedef __attribute__((ext_vector_type(2))) float v2f;
typedef __attribute__((ext_vector_type(8))) float v8f;

// ---------------- wave-synchronous 512-point radix-2 FFT in LDS ------------
// Data must be pre-loaded in bit-reversed order. One wave32 owns the whole
// 512-complex working set; same-wave DS ops retire in-order, so no barrier.
__device__ __forceinline__ void fft512_wave(float* re, float* im, int lane) {
  for (int s = 1; s <= 9; ++s) {
    const int half = 1 << (s - 1);
    #pragma unroll
    for (int it = 0; it < 8; ++it) {
      int j   = it * 32 + lane;                 // butterfly id 0..255
      int pos = j & (half - 1);
      int i0  = ((j >> (s - 1)) << s) + pos;
      int i1  = i0 + half;
      float ang = -3.14159265358979323846f * (float)pos / (float)half;
      float wi, wr;
      __sincosf(ang, &wi, &wr);                 // sin, cos
      float xr = re[i1], xi = im[i1];
      float tr = wr * xr - wi * xi;
      float ti = wr * xi + wi * xr;
      float ur = re[i0], ui = im[i0];
      re[i0] = ur + tr; im[i0] = ui + ti;
      re[i1] = ur - tr; im[i1] = ui - ti;
    }
  }
}

__device__ __forceinline__ float waveRedSum(float v) {
  #pragma unroll
  for (int o = 16; o > 0; o >>= 1) v += __shfl_down(v, o, 32);
  return v;
}
__device__ __forceinline__ float waveRedMax(float v) {
  #pragma unroll
  for (int o = 16; o > 0; o >>= 1) v = fmaxf(v, __shfl_down(v, o, 32));
  return v;
}
__device__ __forceinline__ float waveRedMin(float v) {
  #pragma unroll
  for (int o = 16; o > 0; o >>= 1) v = fminf(v, __shfl_down(v, o, 32));
  return v;
}
__device__ __forceinline__ int waveRedSumI(int v) {
  #pragma unroll
  for (int o = 16; o > 0; o >>= 1) v += __shfl_down(v, o, 32);
  return v;
}

// ---------------- K1: gray + row FFT  (grid 4096 x 256thr, 8 rows/block) ----
__global__ __launch_bounds__(256)
void k_gray_rowfft(const float* __restrict__ x, float* __restrict__ rowfft) {
  __shared__ float re[8][512];
  __shared__ float im[8][512];
  const int wave = threadIdx.x >> 5, lane = threadIdx.x & 31;
  const int rowG = blockIdx.x * 8 + wave;          // 0 .. B*H-1
  const int b = rowG >> 9, h = rowG & 511;
  const float* xr = x + (size_t)b * 3 * HWSZ + (size_t)h * WW;
  const float* xg = xr + HWSZ;
  const float* xb = xg + HWSZ;
  float* R = re[wave]; float* I = im[wave];
  #pragma unroll
  for (int it = 0; it < 16; ++it) {
    int p = it * 32 + lane;                        // coalesced loads
    float g = 0.299f * xr[p] + 0.587f * xg[p] + 0.114f * xb[p];
    int rp = __brev((unsigned)p) >> 23;            // 9-bit bit reverse
    R[rp] = g; I[rp] = 0.0f;
  }
  fft512_wave(R, I, lane);
  float2* out = (float2*)rowfft + ((size_t)b * HH + h) * WW;
  #pragma unroll
  for (int it = 0; it < 16; ++it) {
    int p = it * 32 + lane;
    out[p] = make_float2(R[p], I[p]);              // coalesced 8B stores
  }
}

// ---------------- K2: column FFT + |F| + fftshift + partials ---------------
// grid = B*64 blocks x 256 thr; each block: 8 columns of one image.
__global__ __launch_bounds__(256)
void k_colfft(const float* __restrict__ rowfft, float* __restrict__ mag,
              float* __restrict__ partials) {
  __shared__ float re[8][512];
  __shared__ float im[8][512];
  __shared__ float scr[8][8];
  const int wave = threadIdx.x >> 5, lane = threadIdx.x & 31;
  const int b = blockIdx.x >> 6;                   // image
  const int ctile = blockIdx.x & 63;               // column tile
  const int c0 = ctile * 8;
  const float2* src = (const float2*)rowfft + (size_t)b * HWSZ + c0;

  // cooperative coalesced tile load, bit-reversed row placement
  #pragma unroll
  for (int i = 0; i < 16; ++i) {
    int flat = i * 256 + threadIdx.x;              // 0..4095
    int h = flat >> 3, c = flat & 7;
    float2 v = src[(size_t)h * WW + c];
    int rh = __brev((unsigned)h) >> 23;
    re[c][rh] = v.x; im[c][rh] = v.y;
  }
  __syncthreads();

  fft512_wave(re[wave], im[wave], lane);

  // magnitude (+1e-8 as in reference), overwrite re[] in-place (own wave)
  #pragma unroll
  for (int it = 0; it < 16; ++it) {
    int h = it * 32 + lane;
    float a = re[wave][h], bb = im[wave][h];
    re[wave][h] = sqrtf(a * a + bb * bb) + 1e-8f;
  }
  __syncthreads();

  // cooperative fftshift store + stats
  const int wsC0 = (c0 + 256) & 511;
  float s = 0.f, s2 = 0.f, mx = -1e30f, mn = 1e30f;
  float q0 = 0.f, q1 = 0.f, q2 = 0.f, q3 = 0.f;
  #pragma unroll
  for (int i = 0; i < 16; ++i) {
    int flat = i * 256 + threadIdx.x;
    int h = flat >> 3, c = flat & 7;
    float m = re[c][h];
    int hs = (h + 256) & 511;
    mag[((size_t)b * HH + hs) * WW + wsC0 + c] = m;
    s += m; s2 += m * m;
    mx = fmaxf(mx, m); mn = fminf(mn, m);
    int quad = hs >> 7;
    q0 += (quad == 0) ? m : 0.f;
    q1 += (quad == 1) ? m : 0.f;
    q2 += (quad == 2) ? m : 0.f;
    q3 += (quad == 3) ? m : 0.f;
  }
  // deterministic block reduction (fixed tree order)
  s = waveRedSum(s);  s2 = waveRedSum(s2);
  mx = waveRedMax(mx); mn = waveRedMin(mn);
  q0 = waveRedSum(q0); q1 = waveRedSum(q1);
  q2 = waveRedSum(q2); q3 = waveRedSum(q3);
  if (lane == 0) {
    scr[wave][0] = s;  scr[wave][1] = s2; scr[wave][2] = mx; scr[wave][3] = mn;
    scr[wave][4] = q0; scr[wave][5] = q1; scr[wave][6] = q2; scr[wave][7] = q3;
  }
  __syncthreads();
  if (threadIdx.x < 8) {
    int kind = threadIdx.x;
    float acc = (kind == 2) ? -1e30f : (kind == 3) ? 1e30f : 0.f;
    for (int w = 0; w < 8; ++w) {
      float v = scr[w][kind];
      if (kind == 2)      acc = fmaxf(acc, v);
      else if (kind == 3) acc = fminf(acc, v);
      else                acc += v;
    }
    partials[((size_t)b * 64 + ctile) * 8 + kind] = acc;
  }
}

// ---------------- K3: per-image stats + median + feature vector ------------
// grid = 64 blocks x 256 thr; mag field is L2-resident (67MB < 192MB L2).
__global__ __launch_bounds__(256)
void k_stats(const float* __restrict__ mag, const float* __restrict__ partials,
             float* __restrict__ feats) {
  __shared__ float fin[8];
  __shared__ int   hist[256];
  __shared__ int   icnt[3][8];
  __shared__ float sLo, sHi;
  __shared__ int   sKRem;
  const int b = blockIdx.x;
  const int wave = threadIdx.x >> 5, lane = threadIdx.x & 31;
  const float* mb = mag + (size_t)b * HWSZ;

  if (threadIdx.x < 8) {
    int kind = threadIdx.x;
    float acc = (kind == 2) ? -1e30f : (kind == 3) ? 1e30f : 0.f;
    for (int j = 0; j < 64; ++j) {
      float v = partials[((size_t)b * 64 + j) * 8 + kind];
      if (kind == 2)      acc = fmaxf(acc, v);
      else if (kind == 3) acc = fminf(acc, v);
      else                acc += v;
    }
    fin[kind] = acc;
  }
  __syncthreads();
  const float sum = fin[0], sumsq = fin[1], mxRaw = fin[2], mnRaw = fin[3];
  const float D = mxRaw + 1e-8f;
  const float invD = 1.0f / D;
  const float meanRaw = sum / NTOT;

  // threshold counts (strict >, matching reference)
  int c1 = 0, c2 = 0, c3 = 0;
  const float t2 = 0.5f * D, t3 = 0.1f * D;
  for (int i = 0; i < 1024; ++i) {
    float m = mb[i * 256 + threadIdx.x];
    c1 += (m > meanRaw); c2 += (m > t2); c3 += (m > t3);
  }
  c1 = waveRedSumI(c1); c2 = waveRedSumI(c2); c3 = waveRedSumI(c3);
  if (lane == 0) { icnt[0][wave] = c1; icnt[1][wave] = c2; icnt[2][wave] = c3; }

  // 4-level 256-way histogram refinement for lower median of m = mag/D
  if (threadIdx.x == 0) { sLo = 0.0f; sHi = 1.0f; sKRem = MED_K; }
  __syncthreads();
  for (int lvl = 0; lvl < 4; ++lvl) {
    hist[threadIdx.x] = 0;
    __syncthreads();
    float lo = sLo, hi = sHi;
    float inv = 256.0f / (hi - lo);
    for (int i = 0; i < 1024; ++i) {
      float m = mb[i * 256 + threadIdx.x] * invD;
      if (m >= lo && m < hi) {
        int bi = (int)((m - lo) * inv);
        bi = bi < 255 ? bi : 255;
        atomicAdd(&hist[bi], 1);
      }
    }
    __syncthreads();
    if (threadIdx.x == 0) {
      int acc = 0, kr = sKRem, chosen = 255;
      for (int j = 0; j < 256; ++j) {
        if (acc + hist[j] > kr) { chosen = j; break; }
        acc += hist[j];
      }
      float w = (hi - lo) / 256.0f;
      sKRem = kr - acc;
      sLo = lo + chosen * w;
      sHi = sLo + w;
    }
    __syncthreads();
  }

  if (threadIdx.x == 0) {
    int cg = 0, ch = 0, ct = 0;
    for (int w = 0; w < 8; ++w) { cg += icnt[0][w]; ch += icnt[1][w]; ct += icnt[2][w]; }
    float mean = sum / (NTOT * D);
    float ex2s = sumsq * invD * invD;                // sum of m^2
    float varU = (ex2s - NTOT * mean * mean) / (NTOT - 1.0f);
    float stdv = sqrtf(fmaxf(varU, 0.0f));
    stdv = fmaxf(stdv, 1e-8f);
    float f[12];
    f[0]  = mean;
    f[1]  = stdv;
    f[2]  = mxRaw * invD;
    f[3]  = mnRaw * invD;
    f[4]  = (float)cg / NTOT;
    f[5]  = sLo;                                     // lower median (refined)
    f[6]  = fin[4] / (65536.0f * D);
    f[7]  = fin[5] / (65536.0f * D);
    f[8]  = fin[6] / (65536.0f * D);
    f[9]  = fin[7] / (65536.0f * D);
    f[10] = (float)ch / NTOT;
    f[11] = (float)ct / NTOT;
    #pragma unroll
    for (int i = 0; i < 12; ++i)
      feats[b * 12 + i] = fminf(fmaxf(f[i], -10.0f), 10.0f);
  }
}

// ---------------- K4: MLP with fp32 WMMA (v_wmma_f32_16x16x4_f32) ----------
// One block, 512 threads = 16 wave32s, one 16x16 fp32 tile per wave.
// A 16x4 layout : lanes 0-15 rows M, VGPR0/1 = K0/K1; lanes 16-31 = K2/K3.
// B 4x16 layout : mirror of A (rows K striped, N across lanes).
// C/D 16x16     : VGPR v -> M=v (lanes 0-15) / M=v+8 (lanes 16-31).
__global__ __launch_bounds__(512)
void k_mlp(const float* __restrict__ feats,
           const float* __restrict__ W1, const float* __restrict__ b1,
           const float* __restrict__ g1, const float* __restrict__ be1,
           const float* __restrict__ W2, const float* __restrict__ b2,
           const float* __restrict__ g2, const float* __restrict__ be2,
           const float* __restrict__ W3, const float* __restrict__ b3,
           float* __restrict__ out) {
  __shared__ float A1[64][12];
  __shared__ float H1[64][64];
  __shared__ float H2[64][128];
  const int w = threadIdx.x >> 5, lane = threadIdx.x & 31;
  const int l15 = lane & 15;
  const int koff = (lane >> 4) * 2;
  const int rsel = (lane >> 4) * 8;

  for (int i = threadIdx.x; i < 64 * 12; i += 512)
    A1[i / 12][i % 12] = feats[i];
  __syncthreads();

  // ---- GEMM1: [64x12] @ [12x64]  (16 tiles, 1 per wave) ----
  {
    int mt = w >> 2, nt = w & 3;
    int mrow = mt * 16 + l15;
    int ncol = nt * 16 + l15;
    v8f acc = {};
    for (int k = 0; k < 12; k += 4) {
      v2f a, bf;
      a.x  = A1[mrow][k + koff];
      a.y  = A1[mrow][k + koff + 1];
      bf.x = W1[(k + koff)     * 64 + ncol];
      bf.y = W1[(k + koff + 1) * 64 + ncol];
      acc = __builtin_amdgcn_wmma_f32_16x16x4_f32(false, a, false, bf,
                                                  (short)0, acc, false, false);
    }
    float bias = b1[ncol];
    #pragma unroll
    for (int v = 0; v < 8; ++v)
      H1[mt * 16 + v + rsel][ncol] = acc[v] + bias;
  }
  __syncthreads();

  // ---- BN1 (batch stats over 64 rows, biased var) + ReLU ----
  if (threadIdx.x < 64) {
    int j = threadIdx.x;
    float s = 0.f, s2 = 0.f;
    for (int r = 0; r < 64; ++r) { float v = H1[r][j]; s += v; s2 += v * v; }
    float mu = s / 64.0f;
    float var = s2 / 64.0f - mu * mu;
    float inv = rsqrtf(var + 1e-5f);
    float ga = g1[j], be = be1[j];
    for (int r = 0; r < 64; ++r) {
      float v = ga * (H1[r][j] - mu) * inv + be;
      H1[r][j] = fmaxf(v, 0.0f);
    }
  }
  __syncthreads();

  // ---- GEMM2: [64x64] @ [64x128]  (32 tiles, 2 per wave) ----
  for (int t = 0; t < 2; ++t) {
    int T = w + t * 16;
    int mt = T >> 3, nt = T & 7;
    int mrow = mt * 16 + l15;
    int ncol = nt * 16 + l15;
    v8f acc = {};
    for (int k = 0; k < 64; k += 4) {
      v2f a, bf;
      a.x  = H1[mrow][k + koff];
      a.y  = H1[mrow][k + koff + 1];
      bf.x = W2[(k + koff)     * 128 + ncol];
      bf.y = W2[(k + koff + 1) * 128 + ncol];
      acc = __builtin_amdgcn_wmma_f32_16x16x4_f32(false, a, false, bf,
                                                  (short)0, acc, false, false);
    }
    float bias = b2[ncol];
    #pragma unroll
    for (int v = 0; v < 8; ++v)
      H2[mt * 16 + v + rsel][ncol] = acc[v] + bias;
  }
  __syncthreads();

  // ---- BN2 + ReLU ----
  if (threadIdx.x < 128) {
    int j = threadIdx.x;
    float s = 0.f, s2 = 0.f;
    for (int r = 0; r < 64; ++r) { float v = H2[r][j]; s += v; s2 += v * v; }
    float mu = s / 64.0f;
    float var = s2 / 64.0f - mu * mu;
    float inv = rsqrtf(var + 1e-5f);
    float ga = g2[j], be = be2[j];
    for (int r = 0; r < 64; ++r) {
      float v = ga * (H2[r][j] - mu) * inv + be;
      H2[r][j] = fmaxf(v, 0.0f);
    }
  }
  __syncthreads();

  // ---- GEMM3: [64x128] @ [128x512]  (128 tiles, 8 per wave) ----
  for (int t = 0; t < 8; ++t) {
    int T = w * 8 + t;
    int mt = T >> 5, nt = T & 31;
    int mrow = mt * 16 + l15;
    int ncol = nt * 16 + l15;
    v8f acc = {};
    for (int k = 0; k < 128; k += 4) {
      v2f a, bf;
      a.x  = H2[mrow][k + koff];
      a.y  = H2[mrow][k + koff + 1];
      bf.x = W3[(k + koff)     * 512 + ncol];
      bf.y = W3[(k + koff + 1) * 512 + ncol];
      acc = __builtin_amdgcn_wmma_f32_16x16x4_f32(false, a, false, bf,
                                                  (short)0, acc, false, false);
    }
    float bias = b3[ncol];
    #pragma unroll
    for (int v = 0; v < 8; ++v)
      out[(size_t)(mt * 16 + v + rsel) * 512 + ncol] = acc[v] + bias;
  }
}

// ---------------------------------------------------------------------------
extern "C" void kernel_launch(void* const* d_in, const int* in_sizes, int n_in,
                              void* d_out, int out_size, void* d_ws, size_t ws_size,
                              hipStream_t stream) {
  const float* x   = (const float*)d_in[0];
  const float* W1  = (const float*)d_in[1];
  const float* b1  = (const float*)d_in[2];
  const float* g1  = (const float*)d_in[3];
  const float* be1 = (const float*)d_in[4];
  const float* W2  = (const float*)d_in[5];
  const float* b2  = (const float*)d_in[6];
  const float* g2  = (const float*)d_in[7];
  const float* be2 = (const float*)d_in[8];
  const float* W3  = (const float*)d_in[9];
  const float* b3  = (const float*)d_in[10];
  float* outp = (float*)d_out;

  // workspace layout (~202 MB)
  float* rowfft   = (float*)d_ws;                                  // B*HW*2
  float* mag      = rowfft + (size_t)BATCH * HWSZ * 2;             // B*HW
  float* partials = mag + (size_t)BATCH * HWSZ;                    // B*64*8
  float* feats    = partials + (size_t)BATCH * 64 * 8;             // B*12

  k_gray_rowfft<<<BATCH * HH / 8, 256, 0, stream>>>(x, rowfft);
  k_colfft<<<BATCH * 64, 256, 0, stream>>>(rowfft, mag, partials);
  k_stats<<<BATCH, 256, 0, stream>>>(mag, partials, feats);
  k_mlp<<<1, 512, 0, stream>>>(feats, W1, b1, g1, be1, W2, b2, g2, be2,
                               W3, b3, outp);
}